// DotProductPredictor_65017214927369
// MI455X (gfx1250) — compile-verified
//
#include <hip/hip_runtime.h>
#include <hip/hip_bf16.h>

// Per-edge dot product  score[e] = sum_d h[src[e]][d] * h[dst[e]][d]
// E = 1.6M, D = 128 (f32).  Memory-bound; h (51.2 MB) + edge_index (25.6 MB)
// both fit in MI455X's 192 MB L2, so gathers are L2 hits after first touch.
//
// Strategy: 16 edges per wave.  A = 16 src rows (16xK), B = 16 dst rows
// transposed (Kx16); D = A*B puts the 16 dot products on the diagonal.
// Accumulate over K=128 with 32 chained V_WMMA_F32_16X16X4_F32 ops.
// The f32 WMMA fragment layouts make each lane's per-chunk fragment a single
// contiguous float2 (b64) load:  row_ptr + kc*16B + 8B*(lane>=16)  for BOTH
// the A (src) and B (dst) operands.

typedef float v2f __attribute__((ext_vector_type(2)));
typedef float v8f __attribute__((ext_vector_type(8)));

#define D_FEAT 128

__global__ __launch_bounds__(256) void
u_dot_v_wmma_kernel(const float* __restrict__ h,
                    const long long* __restrict__ edge_index,  // [2, E] int64
                    float* __restrict__ out,                   // [E] f32
                    int n_edges, int n_groups)
{
    const int lane        = threadIdx.x & 31;
    const int wave_in_blk = threadIdx.x >> 5;
    const int group       = blockIdx.x * (blockDim.x >> 5) + wave_in_blk;
    if (group >= n_groups) return;           // wave-uniform: EXEC stays full

    const int base = group * 16;

    // Lane -> edge within the 16-edge group (lanes l and l+16 share an edge).
    int e = base + (lane & 15);
    if (e > n_edges - 1) e = n_edges - 1;    // clamp for tail safety

    const long long s = edge_index[e];                        // src node
    const long long d = edge_index[(long long)n_edges + e];   // dst node

    // Half-wave K-pair select: lanes 0-15 take K={0,1} of each chunk,
    // lanes 16-31 take K={2,3}  -> byte offset 8*(lane>=16).
    const int koff = (lane >> 4) << 1;  // 0 or 2 floats
    const float* rowS = h + (long long)s * D_FEAT + koff;
    const float* rowD = h + (long long)d * D_FEAT + koff;

    v8f acc = {};

    // 32 chunks of K=4; fully unrolled -> immediate offsets (<= 504 B).
#pragma unroll
    for (int kc = 0; kc < 32; ++kc) {
        v2f a = *(const v2f*)(rowS + kc * 4);   // A fragment: M=lane%16
        v2f b = *(const v2f*)(rowD + kc * 4);   // B fragment: N=lane%16
        // 8 args: (neg_a, A, neg_b, B, c_mod, C, reuse_a, reuse_b)
        acc = __builtin_amdgcn_wmma_f32_16x16x4_f32(
            false, a, false, b, (short)0, acc, false, false);
    }

    // Diagonal extraction.  C/D layout: VGPR r holds (M=r, N=lane) on lanes
    // 0-15 and (M=8+r, N=lane-16) on lanes 16-31.  So diag element (i,i):
    //   i in 0..7  -> acc[i]   at lane i
    //   i in 8..15 -> acc[i-8] at lane i+16   (lanes 24..31)
    const int sel = lane & 7;
    float r = acc[0];
    r = (sel == 1) ? acc[1] : r;
    r = (sel == 2) ? acc[2] : r;
    r = (sel == 3) ? acc[3] : r;
    r = (sel == 4) ? acc[4] : r;
    r = (sel == 5) ? acc[5] : r;
    r = (sel == 6) ? acc[6] : r;
    r = (sel == 7) ? acc[7] : r;

    int out_idx = -1;
    if (lane < 8)        out_idx = base + lane;        // edges 0..7
    else if (lane >= 24) out_idx = base + lane - 16;   // edges 8..15
    if (out_idx >= 0 && out_idx < n_edges)
        out[out_idx] = r;
}

extern "C" void kernel_launch(void* const* d_in, const int* in_sizes, int n_in,
                              void* d_out, int out_size, void* d_ws, size_t ws_size,
                              hipStream_t stream) {
    const float*     h  = (const float*)d_in[0];        // [N, 128] f32
    const long long* ei = (const long long*)d_in[1];    // [2, E] int64
    float*           out = (float*)d_out;               // [E, 1] f32

    const int n_edges  = out_size;                      // E
    const int n_groups = (n_edges + 15) / 16;           // 16 edges per wave
    const int waves_per_block = 8;                      // 256 threads (wave32)
    const int blocks = (n_groups + waves_per_block - 1) / waves_per_block;

    u_dot_v_wmma_kernel<<<blocks, 256, 0, stream>>>(h, ei, out, n_edges, n_groups);
}